// TF_NP_MCCD_OPD_34720515621415
// MI455X (gfx1250) — compile-verified
//
#include <hip/hip_runtime.h>
#include <hip/hip_bf16.h>
#include <climits>

typedef __attribute__((ext_vector_type(2))) float v2f;
typedef __attribute__((ext_vector_type(8))) float v8f;

#define N_STARS 4096
#define N_DIC   4096
#define N_FEAT  16
#define OPD     65536   // 256*256 flattened
#define BATCH   2048

// ---------------------------------------------------------------------------
// Kernel 1: faithful tf.where(tf.equal(...)) lookup.
// eq[b,r,c] = positions[b,c]==obs_pos[r,c]; first True in row-major flat
// order; index = flat/2; all-False -> argmax returns 0 -> index 0.
// One 256-thread block per batch element; min-reduce flat index in LDS.
// ---------------------------------------------------------------------------
__global__ void lookup_kernel(const float* __restrict__ positions,
                              const float* __restrict__ obs_pos,
                              int* __restrict__ indices) {
  const int b   = blockIdx.x;
  const int tid = threadIdx.x;
  const float px = positions[2 * b + 0];
  const float py = positions[2 * b + 1];
  int best = INT_MAX;
  for (int flat = tid; flat < 2 * N_STARS; flat += 256) {
    const float v = obs_pos[flat];
    const float p = (flat & 1) ? py : px;
    if (v == p && flat < best) best = flat;
  }
  __shared__ int sbest;
  if (tid == 0) sbest = INT_MAX;
  __syncthreads();
  atomicMin(&sbest, best);
  __syncthreads();
  if (tid == 0) indices[b] = (sbest == INT_MAX) ? 0 : (sbest >> 1);
}

// ---------------------------------------------------------------------------
// Kernel 2: inter[b,f] = sum_k spatial_dic[idx[b],k] * alpha[k,f].
// One block per batch row; 16 accumulators per thread over strided K,
// then LDS tree reduction. 268 MFLOP total - negligible.
// ---------------------------------------------------------------------------
__global__ void inter_kernel(const float* __restrict__ spatial_dic,
                             const float* __restrict__ alpha,
                             const int* __restrict__ indices,
                             float* __restrict__ inter) {
  const int b   = blockIdx.x;
  const int tid = threadIdx.x;
  const float* dic = spatial_dic + (size_t)indices[b] * N_DIC;
  float acc[N_FEAT];
#pragma unroll
  for (int f = 0; f < N_FEAT; ++f) acc[f] = 0.0f;
  for (int k = tid; k < N_DIC; k += 256) {
    const float d = dic[k];
    const float* al = alpha + (size_t)k * N_FEAT;
#pragma unroll
    for (int f = 0; f < N_FEAT; ++f) acc[f] += d * al[f];
  }
  __shared__ float red[256];
#pragma unroll 1
  for (int f = 0; f < N_FEAT; ++f) {
    red[tid] = acc[f];
    __syncthreads();
    for (int s = 128; s > 0; s >>= 1) {
      if (tid < s) red[tid] += red[tid + s];
      __syncthreads();
    }
    if (tid == 0) inter[(size_t)b * N_FEAT + f] = red[0];
    __syncthreads();
  }
}

// ---------------------------------------------------------------------------
// Kernel 3 (dominant, store-bound): out[b,n] = sum_f inter[b,f]*S[f,n].
// [2048,16] x [16,65536] GEMM via V_WMMA_F32_16X16X4_F32, 4 WMMAs per
// 16x16 output tile (K=16). Each wave owns one tile; 8 waves/block cover
// 128 contiguous columns of one 16-row batch tile.
//
// A 16x4 f32 layout (ISA 7.12.2): lane holds M=lane%16; lanes0-15 get
// K={0,1} in VGPR{0,1}, lanes16-31 get K={2,3}. B (4x16) symmetric with
// lane holding N=lane%16. C/D: VGPR r, lanes0-15 -> M=r, lanes16-31 -> M=r+8.
// ---------------------------------------------------------------------------
__global__ void opd_wmma_kernel(const float* __restrict__ inter,
                                const float* __restrict__ S,
                                float* __restrict__ out) {
  const int lane  = threadIdx.x & 31;
  const int wave  = threadIdx.x >> 5;
  const int row16 = lane & 15;
  const int half  = lane >> 4;                       // 0: lanes 0-15, 1: 16-31
  const int btile = blockIdx.y;                      // batch tile (16 rows)
  const int n0    = (blockIdx.x * 8 + wave) * 16;    // column tile base

  const float* arow = inter + (size_t)(btile * 16 + row16) * N_FEAT;

  v2f a[4];
  v2f bm[4];
#pragma unroll
  for (int j = 0; j < 4; ++j) {
    const int k = 4 * j + 2 * half;                  // base K for this lane-half
    a[j].x  = arow[k];
    a[j].y  = arow[k + 1];
    bm[j].x = S[(size_t)k       * OPD + n0 + row16];
    bm[j].y = S[(size_t)(k + 1) * OPD + n0 + row16];
  }

  v8f c = {};
#pragma unroll
  for (int j = 0; j < 4; ++j) {
    c = __builtin_amdgcn_wmma_f32_16x16x4_f32(
        /*neg_a=*/false, a[j], /*neg_b=*/false, bm[j],
        /*c_mod=*/(short)0, c, /*reuse_a=*/false, /*reuse_b=*/false);
  }

#pragma unroll
  for (int r = 0; r < 8; ++r) {
    const size_t m = (size_t)(btile * 16 + r + 8 * half);
    out[m * OPD + (size_t)(n0 + row16)] = c[r];
  }
}

// ---------------------------------------------------------------------------
extern "C" void kernel_launch(void* const* d_in, const int* in_sizes, int n_in,
                              void* d_out, int out_size, void* d_ws, size_t ws_size,
                              hipStream_t stream) {
  const float* positions   = (const float*)d_in[0];  // [2048,2]
  const float* obs_pos     = (const float*)d_in[1];  // [4096,2]
  const float* spatial_dic = (const float*)d_in[2];  // [4096,4096]
  const float* alpha_mat   = (const float*)d_in[3];  // [4096,16]
  const float* S_mat       = (const float*)d_in[4];  // [16,256,256]
  float* out = (float*)d_out;                        // [2048,256,256]

  int*   indices = (int*)d_ws;                                   // 2048 ints
  float* inter   = (float*)((char*)d_ws + BATCH * sizeof(int));  // 2048*16 f32

  lookup_kernel<<<BATCH, 256, 0, stream>>>(positions, obs_pos, indices);
  inter_kernel<<<BATCH, 256, 0, stream>>>(spatial_dic, alpha_mat, indices, inter);

  dim3 grid(OPD / (8 * 16), BATCH / 16);  // (512, 128)
  opd_wmma_kernel<<<grid, 256, 0, stream>>>(inter, S_mat, out);
}